// GraphAttention_1872605741886
// MI455X (gfx1250) — compile-verified
//
#include <hip/hip_runtime.h>
#include <hip/hip_bf16.h>

typedef __attribute__((ext_vector_type(2))) float v2f;
typedef __attribute__((ext_vector_type(8))) float v8f;

#define BB 64
#define NN 512
#define FF 64

// ---------------------------------------------------------------------------
// Tensor Data Mover helper (2D tile, 4-byte elements, optional LDS padding)
// ---------------------------------------------------------------------------
#if __has_builtin(__builtin_amdgcn_tensor_load_to_lds)
#define HAVE_TDM 1
typedef unsigned int u32x4 __attribute__((ext_vector_type(4)));
typedef int i32x4 __attribute__((ext_vector_type(4)));
typedef int i32x8 __attribute__((ext_vector_type(8)));

__device__ __forceinline__ void tdm_load_2d_f32(unsigned lds_byte, const void* gaddr,
                                                unsigned tile_d0, unsigned tile_d1,
                                                unsigned tensor_d0, unsigned tensor_d1,
                                                unsigned stride_d0,
                                                unsigned pad_en,
                                                unsigned pad_interval_code,
                                                unsigned pad_amount_code) {
  unsigned long long ga = (unsigned long long)gaddr;
  u32x4 g0;
  g0[0] = 1u;  // count=1, user mode, no gather
  g0[1] = lds_byte;
  g0[2] = (unsigned)ga;
  g0[3] = (unsigned)((ga >> 32) & 0x01FFFFFFull) | (2u << 30);  // type=2 ("image")
  i32x8 g1;
  g1[0] = (int)((2u << 16) |                  // data_size: 4 bytes
                (pad_en << 20) |              // pad_enable
                (pad_interval_code << 22) |   // pad interval (code)
                (pad_amount_code << 25));     // pad amount (code)
  g1[1] = (int)(tensor_d0 << 16);             // atomic_addr=0 | tensor_dim0[15:0]
  g1[2] = (int)((tensor_d0 >> 16) | (tensor_d1 << 16));
  g1[3] = (int)((tensor_d1 >> 16) | (tile_d0 << 16));
  g1[4] = (int)(tile_d1 & 0xFFFF);            // tile_dim1 | tile_dim2=0
  g1[5] = (int)stride_d0;                     // tensor_dim0_stride[31:0]
  g1[6] = 0;
  g1[7] = 0;
  i32x4 z4 = {0, 0, 0, 0};
#if __clang_major__ >= 23
  i32x8 z8 = {0, 0, 0, 0, 0, 0, 0, 0};
  __builtin_amdgcn_tensor_load_to_lds(g0, g1, z4, z4, z8, 0);
#else
  __builtin_amdgcn_tensor_load_to_lds(g0, g1, z4, z4, 0);
#endif
}
#endif

// ---------------------------------------------------------------------------
// k0: XW = X @ W ; Ytp = pair-interleaved colsum(a)[f]*XW[k,f] (B operand of k1)
//     XWp = pair-interleaved XW (B operand of k2 H-phase)
// ---------------------------------------------------------------------------
__global__ __launch_bounds__(256) void k0_xw(const float* __restrict__ X,
                                             const float* __restrict__ W,
                                             const float* __restrict__ a,
                                             float* __restrict__ XW,
                                             float* __restrict__ Ytp,
                                             float* __restrict__ XWp) {
  __shared__ float sX[64 * 64];
  __shared__ float sW[64 * 64];
  __shared__ float sCS[64];
  const int b = blockIdx.y;
  const int r0 = blockIdx.x * 64;
  const int t = threadIdx.x;

  const float4* Xg = (const float4*)(X + (size_t)b * NN * FF + (size_t)r0 * FF);
  const float4* Wg = (const float4*)W;
  float4* sX4 = (float4*)sX;
  float4* sW4 = (float4*)sW;
#pragma unroll
  for (int i = 0; i < 4; ++i) {
    sX4[t + 256 * i] = Xg[t + 256 * i];
    sW4[t + 256 * i] = Wg[t + 256 * i];
  }
  if (t < 64) {
    float s = 0.f;
    for (int i = 0; i < 64; ++i) s += a[i * 64 + t];
    sCS[t] = s;
  }
  __syncthreads();

  for (int it = 0; it < 16; ++it) {
    const int idx = it * 256 + t;
    const int row = idx >> 6, d = idx & 63;
    float acc = 0.f;
#pragma unroll 8
    for (int c = 0; c < 64; ++c) acc += sX[row * 64 + c] * sW[c * 64 + d];
    const int gr = r0 + row;
    XW[(size_t)b * NN * FF + (size_t)gr * FF + d] = acc;
    Ytp[(size_t)b * FF * NN + (size_t)(d >> 1) * (2 * NN) + gr * 2 + (d & 1)] = sCS[d] * acc;
    XWp[(size_t)b * NN * FF + (size_t)(gr >> 1) * (2 * FF) + d * 2 + (gr & 1)] = acc;
  }
}

// ---------------------------------------------------------------------------
// k1: Zp = pair-interleaved( XW @ Yt + bias_a[k] )   [512,512] per batch
// grid (4 k-tiles x128, 8 i-tiles x64, B); 8 waves: 4(i) x 2(k).
// ---------------------------------------------------------------------------
#define SA1 68    // sXW row stride (conflict-free b64 A-frag reads)
#define SYP 288   // sYtp pair-row stride (256+32; 288 mod 64 == 32)
__global__ __launch_bounds__(256) void k1_z(const float* __restrict__ XW,
                                            const float* __restrict__ Ytp,
                                            const float* __restrict__ bias_a,
                                            float* __restrict__ Zp) {
  __shared__ float sXW[64 * SA1];
  __shared__ float sYtp[32 * SYP];
  const int b = blockIdx.z;
  const int i0 = blockIdx.y * 64;
  const int k0 = blockIdx.x * 128;
  const int t = threadIdx.x;
  const int wave = t >> 5, lane = t & 31, lh = lane >> 4, lm = lane & 15;

  const float* Ytpb = Ytp + (size_t)b * FF * NN;  // 32 pair-rows x 1024 dwords
#if defined(HAVE_TDM)
  if (wave == 0) {
    // issue DMA first so it overlaps the sXW staging below
    tdm_load_2d_f32((unsigned)(size_t)(void*)sYtp, Ytpb + (size_t)k0 * 2,
                    256, 32, 1024, 32, 1024,
                    /*pad_en=*/1, /*interval 256=*/7, /*amount 32=*/31);
  }
#endif
  const float* XWb = XW + (size_t)b * NN * FF + (size_t)i0 * FF;
#pragma unroll
  for (int i = 0; i < 16; ++i) {
    const int idx = t + 256 * i;  // 0..4095
    sXW[(idx >> 6) * SA1 + (idx & 63)] = XWb[idx];
  }
#if defined(HAVE_TDM)
  if (wave == 0) __builtin_amdgcn_s_wait_tensorcnt(0);
#else
#pragma unroll
  for (int i = 0; i < 8; ++i) {
    const int idx = t + 256 * i;  // 2048 float4 units
    const int prow = idx >> 6, dd = (idx & 63) * 4;
    const float4 v = *(const float4*)&Ytpb[(size_t)prow * (2 * NN) + k0 * 2 + dd];
    *(float4*)&sYtp[prow * SYP + dd] = v;
  }
#endif
  __syncthreads();

  const int ib = 16 * (wave >> 1);
  const int kb = 64 * (wave & 1);

  v8f c[4] = {};
  const float* Ab = &sXW[(ib + lm) * SA1 + 2 * lh];
  const float* Bbase = &sYtp[kb * 2 + lm * 2];
  for (int kk = 0; kk < 64; kk += 4) {
    const v2f af = *(const v2f*)&Ab[kk];
    const float* Bb = &Bbase[((kk >> 1) + lh) * SYP];
#pragma unroll
    for (int tt = 0; tt < 4; ++tt) {
      const v2f bf = *(const v2f*)&Bb[32 * tt];
      c[tt] = __builtin_amdgcn_wmma_f32_16x16x4_f32(false, af, false, bf,
                                                    (short)0, c[tt], false, false);
    }
  }

  // store pair-interleaved Z with bias: rows (gi+r, gi+r+1) -> one b64
  float* Zpb = Zp + (size_t)b * NN * NN;
  const int gi = i0 + ib + 8 * lh;  // even
  const int gk0 = k0 + kb + lm;
#pragma unroll
  for (int tt = 0; tt < 4; ++tt) {
    const int gk = gk0 + 16 * tt;
    const float bias = bias_a[gk];
#pragma unroll
    for (int r = 0; r < 8; r += 2) {
      v2f v;
      v.x = c[tt][r] + bias;
      v.y = c[tt][r + 1] + bias;
      *(v2f*)&Zpb[(size_t)((gi + r) >> 1) * (2 * NN) + gk * 2] = v;
    }
  }
}

// ---------------------------------------------------------------------------
// k2: att = tanh(A_ @ Z);  H = att @ XW + bias_W   (fused, 16-row i-tiles)
// grid (32, B), 8 waves; wave w owns att columns [64w, 64w+64).
// Double-buffered TDM pipeline: DMA chunk i+1 overlaps compute of chunk i.
// ---------------------------------------------------------------------------
#define SA2 34     // A_ tile stride
#define SZP 1056   // sZ pair-row stride (1024 + 4*8 TDM pad; 1056 mod 64 == 32)
#define SATT 516   // att tile stride for H-phase A-frag reads
#define NCHUNK 16  // 512 / 32
__global__ __launch_bounds__(256) void k2_att_h(const float* __restrict__ A,
                                                const float* __restrict__ Zp,
                                                const float* __restrict__ XWp,
                                                const long long* __restrict__ Nv,
                                                const float* __restrict__ bias_W,
                                                float* __restrict__ H) {
  __shared__ float sAt[2][16 * SA2];
  __shared__ float sZb[2][16 * SZP];  // buf0 reused as satt(16*516)+sH(16*64)
  const int b = blockIdx.y;
  const int i0 = blockIdx.x * 16;
  const int t = threadIdx.x;
  const int wave = t >> 5, lane = t & 31, lh = lane >> 4, lm = lane & 15;
  const long long nb = Nv[b];
  const float* Ag = A + (size_t)b * NN * NN;
  const float* Zpb = Zp + (size_t)b * NN * NN;
  const int k0w = wave * 64;

#if defined(HAVE_TDM)
  if (wave == 0) {  // prologue: DMA chunk 0 into buf 0
    tdm_load_2d_f32((unsigned)(size_t)(void*)sZb[0], Zpb,
                    1024, 16, 1024, 16, 1024, 1, 7, 7);
  }
#endif

  // per-tt dword offset inside a padded pair-row (never straddles a pad block)
  int dTT[4];
#pragma unroll
  for (int tt = 0; tt < 4; ++tt) {
    const int d0 = (k0w + 16 * tt) * 2;
    dTT[tt] = d0 + 8 * (d0 >> 8) + lm * 2;
  }

  v8f acc[4] = {};

  for (int ic = 0; ic < NCHUNK; ++ic) {
    const int cur = ic & 1, nxt = cur ^ 1;
    const int jc = ic * 32;
    // stage A_ tile 16x32 (+1 on diagonal where i < N[b])
    {
      int idx = t;
#pragma unroll
      for (int s = 0; s < 2; ++s, idx += 256) {
        const int r = idx >> 5, cc = idx & 31;
        const int gi = i0 + r, gj = jc + cc;
        float v = Ag[(size_t)gi * NN + gj];
        if (gi == gj && (long long)gi < nb) v += 1.0f;
        sAt[cur][r * SA2 + cc] = v;
      }
    }
#if defined(HAVE_TDM)
    if (wave == 0) {
      if (ic < NCHUNK - 1) {  // prefetch chunk ic+1, then wait only for chunk ic
        tdm_load_2d_f32((unsigned)(size_t)(void*)sZb[nxt],
                        Zpb + (size_t)(jc + 32) * NN,
                        1024, 16, 1024, 16, 1024, 1, 7, 7);
        __builtin_amdgcn_s_wait_tensorcnt(1);
      } else {
        __builtin_amdgcn_s_wait_tensorcnt(0);
      }
    }
#else
#pragma unroll
    for (int s = 0; s < 16; ++s) {
      const int idx = t + 256 * s;  // 4096 float4 units
      const int prow = idx >> 8, dd = (idx & 255) * 4;
      const float4 v = *(const float4*)&Zpb[(size_t)((jc >> 1) + prow) * (2 * NN) + dd];
      *(float4*)&sZb[cur][prow * SZP + dd + 8 * (dd >> 8)] = v;
    }
#endif
    __syncthreads();

    const float* Ab = &sAt[cur][lm * SA2 + 2 * lh];
    const float* Zcur = sZb[cur];
    for (int kk = 0; kk < 32; kk += 4) {
      const v2f af = *(const v2f*)&Ab[kk];
      const float* Bb = &Zcur[((kk >> 1) + lh) * SZP];
#pragma unroll
      for (int tt = 0; tt < 4; ++tt) {
        const v2f bf = *(const v2f*)&Bb[dTT[tt]];
        acc[tt] = __builtin_amdgcn_wmma_f32_16x16x4_f32(false, af, false, bf,
                                                        (short)0, acc[tt], false, false);
      }
    }
    __syncthreads();
  }

  // tanh in registers
#pragma unroll
  for (int tt = 0; tt < 4; ++tt)
#pragma unroll
    for (int r = 0; r < 8; ++r) acc[tt][r] = tanhf(acc[tt][r]);

  // spill att tile (16x512) to LDS over buf0 (last compute chunk used buf1)
  float* satt = sZb[0];
  float* sH = sZb[0] + 16 * SATT;
#pragma unroll
  for (int tt = 0; tt < 4; ++tt) {
    const int n = k0w + 16 * tt + lm;
#pragma unroll
    for (int r = 0; r < 8; ++r) satt[(8 * lh + r) * SATT + n] = acc[tt][r];
  }
#pragma unroll
  for (int s = 0; s < 4; ++s) sH[t + 256 * s] = 0.f;
  __syncthreads();

  // H partial: att[i, k0w:k0w+64] @ XW[k0w:k0w+64, :]; B-frags = b64 from XWp
  v8f h[4] = {};
  const float* XWpb = XWp + (size_t)b * NN * FF;
  const float* Atb = &satt[lm * SATT + k0w + 2 * lh];
  for (int kk = 0; kk < 64; kk += 4) {
    const v2f af = *(const v2f*)&Atb[kk];
    const float* Bg = &XWpb[(size_t)(((k0w + kk) >> 1) + lh) * (2 * FF) + lm * 2];
#pragma unroll
    for (int tt = 0; tt < 4; ++tt) {
      const v2f bf = *(const v2f*)&Bg[32 * tt];
      h[tt] = __builtin_amdgcn_wmma_f32_16x16x4_f32(false, af, false, bf,
                                                    (short)0, h[tt], false, false);
    }
  }
  // reduce the 8 per-wave K-partials
#pragma unroll
  for (int tt = 0; tt < 4; ++tt) {
    const int n = 16 * tt + lm;
#pragma unroll
    for (int r = 0; r < 8; ++r) atomicAdd(&sH[(8 * lh + r) * FF + n], h[tt][r]);
  }
  __syncthreads();

  float* Hb = H + (size_t)b * NN * FF + (size_t)i0 * FF;
#pragma unroll
  for (int s = 0; s < 4; ++s) {
    const int idx = t + 256 * s;  // 0..1023
    Hb[idx] = sH[idx] + bias_W[idx & 63];
  }
}

// ---------------------------------------------------------------------------
extern "C" void kernel_launch(void* const* d_in, const int* in_sizes, int n_in,
                              void* d_out, int out_size, void* d_ws, size_t ws_size,
                              hipStream_t stream) {
  (void)in_sizes; (void)n_in; (void)out_size; (void)ws_size;
  const float* X = (const float*)d_in[0];
  const float* A = (const float*)d_in[1];
  const long long* Nv = (const long long*)d_in[2];
  const float* W = (const float*)d_in[3];
  const float* a = (const float*)d_in[4];
  const float* bias_W = (const float*)d_in[5];
  const float* bias_a = (const float*)d_in[6];
  float* H = (float*)d_out;

  float* XW = (float*)d_ws;                    //  8 MB
  float* Ytp = XW + (size_t)BB * NN * FF;      //  8 MB (pair-interleaved)
  float* XWp = Ytp + (size_t)BB * FF * NN;     //  8 MB (pair-interleaved)
  float* Zp = XWp + (size_t)BB * NN * FF;      // 64 MB (pair-interleaved)

  k0_xw<<<dim3(8, BB), 256, 0, stream>>>(X, W, a, XW, Ytp, XWp);
  k1_z<<<dim3(4, 8, BB), 256, 0, stream>>>(XW, Ytp, bias_a, Zp);
  k2_att_h<<<dim3(32, BB), 256, 0, stream>>>(A, Zp, XWp, Nv, bias_W, H);
}